// SpecNet_55439437857185
// MI455X (gfx1250) — compile-verified
//
#include <hip/hip_runtime.h>

typedef float v2f __attribute__((ext_vector_type(2)));
typedef float v8f __attribute__((ext_vector_type(8)));

#define BATCH  32
#define NODES  20000
#define FEAT   64     // i
#define OUTJ   64     // j
#define EIG    16     // e / f
#define CHUNK  500    // nodes per pass-1 chunk (multiple of 4)
#define NCHUNK 40     // NODES / CHUNK
#define NTILE  1250   // NODES / 16

// ---------------------------------------------------------------------------
// Pass 1: partial Z tiles.  Z_b = X_b^T (64 x 20000)  *  V (20000 x 16)
// One block = (b, chunk); 4 waves, wave w owns i-tile [16w, 16w+16).
// A tile (16x4 fp32, ISA layout): lane(m,hk): VGPR0 <-> K=2*hk, VGPR1 <-> K=2*hk+1
// B tile (4x16 fp32): mirrored: VGPR0 row K=2*hk, VGPR1 row K=2*hk+1, N = m
// ---------------------------------------------------------------------------
__global__ __launch_bounds__(128) void spec_z_partial(
    const float* __restrict__ x, const float* __restrict__ V,
    float* __restrict__ partial)
{
  const int chunk = blockIdx.x;
  const int b     = blockIdx.y;
  const int wave  = threadIdx.x >> 5;
  const int lane  = threadIdx.x & 31;
  const int m     = lane & 15;
  const int hk    = lane >> 4;
  const int i0    = wave * 16;
  const int n0    = chunk * CHUNK;
  const float* xb = x + (size_t)b * NODES * FEAT;

  v8f acc = {};
  for (int k = 0; k < CHUNK; k += 4) {
    const int n = n0 + k + 2 * hk;
    v2f a, bb;
    // A[M=m][K=2hk], A[M=m][K=2hk+1]  (A = X^T  =>  x[b][n][i0+m])
    a.x  = __builtin_nontemporal_load(&xb[(size_t)n       * FEAT + i0 + m]);
    a.y  = __builtin_nontemporal_load(&xb[(size_t)(n + 1) * FEAT + i0 + m]);
    // B[K=2hk][N=m], B[K=2hk+1][N=m]  (B = V rows)
    bb.x = V[n * EIG + m];
    bb.y = V[(n + 1) * EIG + m];
    acc = __builtin_amdgcn_wmma_f32_16x16x4_f32(
        /*neg_a=*/false, a, /*neg_b=*/false, bb,
        /*c_mod=*/(short)0, acc, /*reuse_a=*/false, /*reuse_b=*/false);
  }

  // C layout: VGPR r -> M = r (lanes 0-15) / r+8 (lanes 16-31), N = m
  float* p = partial + ((size_t)b * NCHUNK + chunk) * (FEAT * EIG);
#pragma unroll
  for (int r = 0; r < 8; ++r) {
    const int i = i0 + r + 8 * hk;
    p[i * EIG + m] = acc[r];
  }
}

// ---------------------------------------------------------------------------
// Pass 1b: deterministic fixed-order reduction of the NCHUNK partials -> z
// ---------------------------------------------------------------------------
__global__ __launch_bounds__(256) void spec_z_reduce(
    const float* __restrict__ partial, float* __restrict__ z)
{
  const int idx = blockIdx.x * 256 + threadIdx.x;   // 0 .. 32767
  const int b   = idx >> 10;
  const int rem = idx & 1023;                       // i*16 + f
  const float* p = partial + (size_t)b * NCHUNK * 1024 + rem;
  float s = 0.f;
#pragma unroll 4
  for (int c = 0; c < NCHUNK; ++c) s += p[(size_t)c * 1024];
  z[idx] = s;
}

// ---------------------------------------------------------------------------
// Pass 2 (tiny, 67 MFLOP): wT[b][e][j] = sum_{i,f} G[j][i][e][f] * z[b][i][f]
// One block per (j, b); 64 threads: t = e*4 + quarter(i).
// Stored transposed (e major) so pass 3's B operand is row-contiguous.
// ---------------------------------------------------------------------------
__global__ __launch_bounds__(64) void spec_filter(
    const float* __restrict__ G, const float* __restrict__ z,
    float* __restrict__ wT)
{
  __shared__ float zs[FEAT * EIG];   // 4 KB
  __shared__ float red[64];
  const int j   = blockIdx.x;
  const int b   = blockIdx.y;
  const int tid = threadIdx.x;

  for (int t = tid; t < FEAT * EIG; t += 64)
    zs[t] = z[(size_t)b * FEAT * EIG + t];
  __syncthreads();

  const int e = tid >> 2;
  const int q = tid & 3;
  const float* g = G + (size_t)j * FEAT * EIG * EIG;
  float s = 0.f;
  for (int ii = 0; ii < 16; ++ii) {
    const int i = q * 16 + ii;
    const float* gr = g + ((size_t)i * EIG + e) * EIG;
#pragma unroll
    for (int f = 0; f < EIG; ++f) s += gr[f] * zs[i * EIG + f];
  }
  red[tid] = s;
  __syncthreads();
  if (tid < 16) {
    float v = red[tid * 4] + red[tid * 4 + 1] + red[tid * 4 + 2] + red[tid * 4 + 3];
    wT[((size_t)b * EIG + tid) * OUTJ + j] = v;
  }
}

// ---------------------------------------------------------------------------
// Pass 3: out_b[n0:n0+16, :] = V[n0:n0+16, :] (16x16e) * wT_b (16e x 64j)
// One wave per (ntile, b): 4 j-tiles x 4 K-steps = 16 WMMAs; NT stores.
// ---------------------------------------------------------------------------
__global__ __launch_bounds__(32) void spec_out(
    const float* __restrict__ V, const float* __restrict__ wT,
    float* __restrict__ out)
{
  const int nt   = blockIdx.x;
  const int b    = blockIdx.y;
  const int lane = threadIdx.x & 31;
  const int m    = lane & 15;
  const int hk   = lane >> 4;
  const int n0   = nt * 16;
  const float* wb = wT + (size_t)b * EIG * OUTJ;

  v8f acc[4] = {};

#pragma unroll
  for (int kk = 0; kk < 4; ++kk) {
    const int e0 = kk * 4 + 2 * hk;
    v2f a;  // A[M=m][K=e0], A[M=m][K=e0+1]  (A = V rows)
    a.x = V[(size_t)(n0 + m) * EIG + e0];
    a.y = V[(size_t)(n0 + m) * EIG + e0 + 1];
#pragma unroll
    for (int jt = 0; jt < 4; ++jt) {
      v2f bb;  // B[K=e0][N=jt*16+m], B[K=e0+1][N=jt*16+m]
      bb.x = wb[(size_t)e0 * OUTJ + jt * 16 + m];
      bb.y = wb[(size_t)(e0 + 1) * OUTJ + jt * 16 + m];
      acc[jt] = __builtin_amdgcn_wmma_f32_16x16x4_f32(
          false, a, false, bb, (short)0, acc[jt], false, false);
    }
  }

  float* ob = out + ((size_t)b * NODES + n0) * OUTJ;
#pragma unroll
  for (int jt = 0; jt < 4; ++jt) {
#pragma unroll
    for (int r = 0; r < 8; ++r) {
      __builtin_nontemporal_store(acc[jt][r],
          &ob[(size_t)(r + 8 * hk) * OUTJ + jt * 16 + m]);
    }
  }
}

// ---------------------------------------------------------------------------
extern "C" void kernel_launch(void* const* d_in, const int* in_sizes, int n_in,
                              void* d_out, int out_size, void* d_ws, size_t ws_size,
                              hipStream_t stream)
{
  const float* x = (const float*)d_in[0];   // (32, 20000, 64)
  const float* V = (const float*)d_in[1];   // (20000, 16)
  const float* G = (const float*)d_in[2];   // (64, 64, 16, 16)
  float* out = (float*)d_out;               // (32, 20000, 64)

  // workspace: 5.24 MB partials + 128 KB z + 128 KB wT
  float* partial = (float*)d_ws;                                   // B*NCHUNK*64*16
  float* z  = partial + (size_t)BATCH * NCHUNK * FEAT * EIG;       // B*64*16
  float* wTp = z + (size_t)BATCH * FEAT * EIG;                     // B*16*64

  spec_z_partial<<<dim3(NCHUNK, BATCH), 128, 0, stream>>>(x, V, partial);
  spec_z_reduce <<<dim3((BATCH * FEAT * EIG) / 256), 256, 0, stream>>>(partial, z);
  spec_filter   <<<dim3(OUTJ, BATCH), 64, 0, stream>>>(G, z, wTp);
  spec_out      <<<dim3(NTILE, BATCH), 32, 0, stream>>>(V, wTp, out);
}